// MultiHeadAttentionLayerGritSparse_90512140796687
// MI455X (gfx1250) — compile-verified
//
#include <hip/hip_runtime.h>

#define NN   50000        // nodes
#define NE   800000       // edges
#define IND  64           // input dim
#define HD   64           // H*D
#define NH   8            // heads
#define ECOL 128          // 2*H*D

typedef __attribute__((ext_vector_type(2))) float v2f;
typedef __attribute__((ext_vector_type(4))) float v4f;
typedef __attribute__((ext_vector_type(8))) float v8f;

__device__ __forceinline__ v8f wmma_f32(v2f a, v2f b, v8f c) {
    // V_WMMA_F32_16X16X4_F32 : D = A(16x4) x B(4x16) + C(16x16), all fp32
    return __builtin_amdgcn_wmma_f32_16x16x4_f32(
        /*neg_a=*/false, a, /*neg_b=*/false, b,
        /*c_mod=*/(short)0, c, /*reuse_a=*/false, /*reuse_b=*/false);
}

__global__ void k_zero(float* __restrict__ p, int n) {
    int i = blockIdx.x * blockDim.x + threadIdx.x;
    if (i < n) p[i] = 0.f;
}

// ------- Kernel 0: transpose weights so B fragments are contiguous b64 loads -------
// WqkvT: [192][64]  rows 0-63 = Qw^T, 64-127 = Kw^T, 128-191 = Vw^T
// EwT:   [128][64]  = Ew^T
__global__ void k_transpose(const float* __restrict__ Qw, const float* __restrict__ Kw,
                            const float* __restrict__ Vw, const float* __restrict__ Ew,
                            float* __restrict__ WqkvT, float* __restrict__ EwT)
{
    const int ti = blockIdx.x * blockDim.x + threadIdx.x;
    if (ti < 192 * 64) {
        const int r = ti >> 6, k = ti & 63;
        float v;
        if (r < 64)       v = Qw[k * HD + r];
        else if (r < 128) v = Kw[k * HD + (r - 64)];
        else              v = Vw[k * HD + (r - 128)];
        WqkvT[ti] = v;
    }
    const int t2 = ti - 192 * 64;
    if (t2 >= 0 && t2 < 128 * 64) {
        const int r = t2 >> 6, k = t2 & 63;
        EwT[t2] = Ew[k * ECOL + r];
    }
}

// ---------------- Kernel 1: QKV node GEMMs (WMMA fp32) ----------------
// One wave per (16-node tile, 16-col tile). 12 col tiles = Q(4)+K(4)+V(4).
__global__ void k_qkv(const float* __restrict__ x,
                      const float* __restrict__ WqkvT, const float* __restrict__ Qb,
                      float* __restrict__ Qbuf, float* __restrict__ Kbuf,
                      float* __restrict__ Vbuf)
{
    const int wave = blockIdx.x * (blockDim.x >> 5) + (threadIdx.x >> 5);
    const int NWAVES = (NN / 16) * 12;
    if (wave >= NWAVES) return;                 // wave-uniform guard (EXEC all-1 for WMMA)
    const int mtile = wave / 12;
    const int ct    = wave % 12;                // 0-3: Q, 4-7: K, 8-11: V
    const int lane  = threadIdx.x & 31;
    const int half  = lane >> 4;                // 0: lanes 0-15, 1: lanes 16-31
    const int lm    = lane & 15;

    float* outp;
    int c0;
    if (ct < 4)      { outp = Qbuf; c0 = ct * 16; }
    else if (ct < 8) { outp = Kbuf; c0 = (ct - 4) * 16; }
    else             { outp = Vbuf; c0 = (ct - 8) * 16; }

    // Fixed per-lane bases; unrolled kk becomes immediate offsets.
    const float* __restrict__ ap = x + (mtile * 16 + lm) * IND + half * 2;
    const float* __restrict__ bp = WqkvT + (ct * 16 + lm) * IND + half * 2;

    v8f acc = {0.f,0.f,0.f,0.f,0.f,0.f,0.f,0.f};
    #pragma unroll
    for (int kk = 0; kk < IND; kk += 4) {
        const v2f a = *(const v2f*)(ap + kk);   // contiguous b64: A[M][ka..ka+1]
        const v2f b = *(const v2f*)(bp + kk);   // contiguous b64: W^T[c][ka..ka+1]
        acc = wmma_f32(a, b, acc);
    }
    const float bias = (ct < 4) ? Qb[c0 + lm] : 0.f;
    #pragma unroll
    for (int j = 0; j < 8; ++j)                 // C/D: row = j + half*8, col = lm
        outp[(mtile * 16 + j + half * 8) * HD + c0 + lm] = acc[j] + bias;
}

// ------- Kernel 2: fused edge GEMM + gate + logits + softmax denom -------
// One wave per 16-edge tile. 8 fp32 accumulators span all 128 E columns,
// so the [NE,128] E intermediate never hits HBM.
__global__ void k_edge(const float* __restrict__ ea,
                       const int* __restrict__ srcI, const int* __restrict__ dstI,
                       const float* __restrict__ EwT, const float* __restrict__ Eb,
                       const float* __restrict__ Qbuf, const float* __restrict__ Kbuf,
                       const float* __restrict__ Aw,
                       float* __restrict__ wE, float* __restrict__ expv,
                       float* __restrict__ zbuf)
{
    const int wave = blockIdx.x * (blockDim.x >> 5) + (threadIdx.x >> 5);
    if (wave >= NE / 16) return;
    const int e0   = wave * 16;
    const int lane = threadIdx.x & 31;
    const int half = lane >> 4;
    const int lm   = lane & 15;

    const float* __restrict__ ap = ea + (size_t)(e0 + lm) * IND + half * 2;
    const float* __restrict__ bp = EwT + lm * IND + half * 2;   // + t*16*IND per tile

    v8f acc[8];
    #pragma unroll
    for (int t = 0; t < 8; ++t) acc[t] = v8f{0.f,0.f,0.f,0.f,0.f,0.f,0.f,0.f};

    #pragma unroll
    for (int kk = 0; kk < IND; kk += 4) {
        const v2f a = *(const v2f*)(ap + kk);
        #pragma unroll
        for (int t = 0; t < 8; ++t) {
            const v2f b = *(const v2f*)(bp + t * 16 * IND + kk);
            acc[t] = wmma_f32(a, b, acc[t]);
        }
    }
    // add Eb bias (column-only)
    #pragma unroll
    for (int t = 0; t < 8; ++t) {
        const float eb = Eb[t * 16 + lm];
        #pragma unroll
        for (int j = 0; j < 8; ++j) acc[t][j] += eb;
    }

    // Epilogue: gate, wE store, per-head logit reduction, exp + atomic denom.
    // Tile t in [0,4) holds E_w cols, t+4 holds matching E_b cols.
    const int d = lm & 7;                       // per-head feature index
    #pragma unroll
    for (int j = 0; j < 8; ++j) {
        const int r  = e0 + j + half * 8;       // global edge row of this C element
        const int s  = srcI[r];
        const int dn = dstI[r];
        #pragma unroll
        for (int t = 0; t < 4; ++t) {
            const int c = t * 16 + lm;          // column in [0,64) -> (h = c>>3, d = c&7)
            const float kq = Kbuf[s * HD + c] + Qbuf[dn * HD + c];
            float w = fmaxf(kq * acc[t][j] + acc[t + 4][j], 0.f);
            wE[(size_t)r * HD + c] = w;
            const int h = c >> 3;
            float p = w * Aw[d * NH + h];       // Aw is (D,H,1): Aw[d][h]
            p += __shfl_xor(p, 1, 32);          // butterfly over the 8 lanes of one head
            p += __shfl_xor(p, 2, 32);
            p += __shfl_xor(p, 4, 32);
            if ((lane & 7) == 0) {
                const float lg = fminf(fmaxf(p, -5.f), 5.f);
                const float ev = __expf(lg);    // clamp => exp(logit) safe, no segment-max
                expv[r * NH + h] = ev;
                atomicAdd(&zbuf[dn * NH + h], ev);
            }
        }
    }
}

// -------- Kernel 3: normalize + weighted message scatter-add --------
// One thread per (edge, head): one index/denominator fetch, b128 vector loads,
// 16 native global_atomic_add_f32.
__global__ void k_scatter(const int* __restrict__ srcI, const int* __restrict__ dstI,
                          const float* __restrict__ Vbuf, const float* __restrict__ wE,
                          const float* __restrict__ expv, const float* __restrict__ zbuf,
                          float* __restrict__ wV, float* __restrict__ rowV)
{
    const int ti = blockIdx.x * blockDim.x + threadIdx.x;
    if (ti >= NE * NH) return;
    const int e = ti >> 3;
    const int h = ti & 7;
    const int s = srcI[e], dn = dstI[e];
    const float a = expv[ti] / (zbuf[dn * NH + h] + 1e-16f);

    const v4f* vp = (const v4f*)(Vbuf + (size_t)s * HD + h * 8);
    const v4f* wp = (const v4f*)(wE + (size_t)e * HD + h * 8);
    const v4f v0 = vp[0], v1 = vp[1];
    const v4f w0 = wp[0], w1 = wp[1];

    float* wVp = wV   + (size_t)dn * HD + h * 8;
    float* rVp = rowV + (size_t)dn * HD + h * 8;
    #pragma unroll
    for (int dd = 0; dd < 4; ++dd) {
        atomicAdd(&wVp[dd],     v0[dd] * a);
        atomicAdd(&wVp[dd + 4], v1[dd] * a);
        atomicAdd(&rVp[dd],     w0[dd] * a);
        atomicAdd(&rVp[dd + 4], w1[dd] * a);
    }
}

// -------- Kernel 4: wV += rowV @ VeRow (per-head 8x8, tiny) --------
__global__ void k_final(const float* __restrict__ rowV, const float* __restrict__ VeRow,
                        float* __restrict__ wV)
{
    const int ti = blockIdx.x * blockDim.x + threadIdx.x;
    if (ti >= NN * HD) return;
    const int n = ti >> 6;
    const int c = ti & 63;
    const int h = c >> 3, dc = c & 7;
    const float* rp = rowV + n * HD + h * 8;
    float sum = 0.f;
    #pragma unroll
    for (int dd = 0; dd < 8; ++dd)              // VeRow is (D,H,D): VeRow[dd][h][dc]
        sum += rp[dd] * VeRow[dd * HD + h * 8 + dc];
    wV[ti] += sum;
}

extern "C" void kernel_launch(void* const* d_in, const int* in_sizes, int n_in,
                              void* d_out, int out_size, void* d_ws, size_t ws_size,
                              hipStream_t stream) {
    const float* x     = (const float*)d_in[0];
    const float* ea    = (const float*)d_in[1];
    const int*   ei    = (const int*)  d_in[2];
    const float* Qw    = (const float*)d_in[3];
    const float* Qb    = (const float*)d_in[4];
    const float* Kw    = (const float*)d_in[5];
    const float* Ew    = (const float*)d_in[6];
    const float* Eb    = (const float*)d_in[7];
    const float* Vw    = (const float*)d_in[8];
    const float* Aw    = (const float*)d_in[9];
    const float* VeRow = (const float*)d_in[10];

    float* out = (float*)d_out;
    float* wV  = out;                               // [NN, 64]
    float* wE  = out + (size_t)NN * HD;             // [NE, 64]

    float* ws    = (float*)d_ws;                    // ~78.5 MB used
    float* Qbuf  = ws;
    float* Kbuf  = Qbuf + (size_t)NN * HD;
    float* Vbuf  = Kbuf + (size_t)NN * HD;
    float* expv  = Vbuf + (size_t)NN * HD;          // [NE, 8]
    float* zbuf  = expv + (size_t)NE * NH;          // [NN, 8]
    float* rowV  = zbuf + (size_t)NN * NH;          // [NN, 64]
    float* WqkvT = rowV + (size_t)NN * HD;          // [192, 64]
    float* EwT   = WqkvT + 192 * 64;                // [128, 64]

    const int* srcI = ei;
    const int* dstI = ei + NE;

    k_zero<<<(NN * HD + 255) / 256, 256, 0, stream>>>(wV,   NN * HD);
    k_zero<<<(NN * NH + 255) / 256, 256, 0, stream>>>(zbuf, NN * NH);
    k_zero<<<(NN * HD + 255) / 256, 256, 0, stream>>>(rowV, NN * HD);

    k_transpose<<<(192 * 64 + 128 * 64 + 255) / 256, 256, 0, stream>>>(
        Qw, Kw, Vw, Ew, WqkvT, EwT);

    const int waves1 = (NN / 16) * 12;              // 37500 waves
    k_qkv<<<(waves1 + 7) / 8, 256, 0, stream>>>(x, WqkvT, Qb, Qbuf, Kbuf, Vbuf);

    const int waves2 = NE / 16;                     // 50000 waves
    k_edge<<<(waves2 + 7) / 8, 256, 0, stream>>>(ea, srcI, dstI, EwT, Eb,
                                                 Qbuf, Kbuf, Aw, wE, expv, zbuf);

    k_scatter<<<(NE * NH + 255) / 256, 256, 0, stream>>>(srcI, dstI, Vbuf, wE,
                                                         expv, zbuf, wV, rowV);

    k_final<<<(NN * HD + 255) / 256, 256, 0, stream>>>(rowV, VeRow, wV);
}